// CombinedLoss_39256001086051
// MI455X (gfx1250) — compile-verified
//
#include <hip/hip_runtime.h>
#include <math.h>

typedef __attribute__((ext_vector_type(2))) float v2f;
typedef __attribute__((ext_vector_type(8))) float v8f;

#define NPTS  2048
#define BATCH 8
#define KNORM 16
#define KREP  4

// ---------------------------------------------------------------------------
// CDNA5 async global->LDS staging (ASYNCcnt path, bypasses VGPRs).
// ---------------------------------------------------------------------------
__device__ inline void async_copy_b128(unsigned lds_addr, unsigned long long gaddr) {
    asm volatile("global_load_async_to_lds_b128 %0, %1, off"
                 :: "v"(lds_addr), "v"(gaddr) : "memory");
}
__device__ inline void wait_async_all() {
    asm volatile("s_wait_asynccnt 0x0" ::: "memory");
}

// ---------------------------------------------------------------------------
// init: colmin = +inf (uint bits), acc = 0
// ---------------------------------------------------------------------------
__global__ void init_kernel(unsigned int* __restrict__ colmin,
                            float* __restrict__ acc) {
    int i = blockIdx.x * blockDim.x + threadIdx.x;
    if (i < BATCH * NPTS) colmin[i] = 0x7F800000u;   // +inf
    if (i < 8) acc[i] = 0.0f;
}

// ---------------------------------------------------------------------------
// Chamfer via V_WMMA_F32_16X16X4_F32:
//   D = xx + yy - 2 * (pred . gt^T), K=3 padded to 4 with zeros.
// 256-thread block = 8 waves; gt cloud (24 KB) async-staged into LDS once
// per block; each wave owns TWO 16-row pred tiles and reuses each B fragment
// for two back-to-back WMMAs (double matrix-op density per tile overhead).
// A layout (32-bit 16x4): lanes 0-15 -> M=lane, K=0,1 ; lanes 16-31 -> K=2,3.
// C/D layout: element (M = v + 8*(lane>=16), N = lane%16) in VGPR v.
// ---------------------------------------------------------------------------
__global__ __launch_bounds__(256) void chamfer_wmma_kernel(
        const float* __restrict__ pred, const float* __restrict__ gt,
        unsigned int* __restrict__ colmin, float* __restrict__ acc) {
    __shared__ float sgt[NPTS * 3];           // 24 KB of the 320 KB WGP LDS

    const int b    = blockIdx.x >> 3;         // batch
    const int rt0  = (blockIdx.x & 7) * 16;   // first of 16 row tiles
    const int tid  = threadIdx.x;
    const int wave = tid >> 5;
    const int lane = tid & 31;
    const int half = lane >> 4;
    const int lr   = lane & 15;

    const float* pb = pred + (size_t)b * NPTS * 3;
    const float* gb = gt   + (size_t)b * NPTS * 3;

    // ---- async-stage gt cloud into LDS: 24576 B = 6 x (256 lanes x 16 B) ----
    const unsigned lds_base = (unsigned)(unsigned long long)(void*)sgt;
#pragma unroll
    for (int it = 0; it < (NPTS * 3 * 4) / (256 * 16); ++it) {
        const int off = (it * 256 + tid) * 16;
        async_copy_b128(lds_base + (unsigned)off,
                        (unsigned long long)((const char*)gb + off));
    }
    wait_async_all();
    __syncthreads();

    // ---- two A matrices (loop invariant): row tiles r0 and r1 ----
    const int r0 = (rt0 + wave * 2) * 16;
    const int r1 = r0 + 16;

    const float* pr0 = pb + (size_t)(r0 + lr) * 3;
    const float p0x = pr0[0], p0y = pr0[1], p0z = pr0[2];
    v2f A0; A0.x = half ? p0z : p0x; A0.y = half ? 0.0f : p0y;
    const float xx0_lr = p0x * p0x + p0y * p0y + p0z * p0z;

    const float* pr1 = pb + (size_t)(r1 + lr) * 3;
    const float p1x = pr1[0], p1y = pr1[1], p1z = pr1[2];
    v2f A1; A1.x = half ? p1z : p1x; A1.y = half ? 0.0f : p1y;
    const float xx1_lr = p1x * p1x + p1y * p1y + p1z * p1z;

    // each lane needs xx[v + 8*half] (accumulator row mapping)
    float xxv0[8], xxv1[8];
#pragma unroll
    for (int v = 0; v < 8; ++v) {
        xxv0[v] = __shfl(xx0_lr, v + 8 * half, 32);
        xxv1[v] = __shfl(xx1_lr, v + 8 * half, 32);
    }

    float rowmin0[8], rowmin1[8];
#pragma unroll
    for (int v = 0; v < 8; ++v) { rowmin0[v] = 3.0e38f; rowmin1[v] = 3.0e38f; }

    for (int t = 0; t < NPTS / 16; ++t) {
        // ---- B fragment: gt columns t*16 .. t*16+15, served from LDS ----
        const float* gr = &sgt[(t * 16 + lr) * 3];
        const float gx = gr[0], gy = gr[1], gz = gr[2];
        v2f Bm; Bm.x = half ? gz : gx; Bm.y = half ? 0.0f : gy;
        const float yy_col = gx * gx + gy * gy + gz * gz; // col (t*16+lr)

        v8f c0 = {};
        c0 = __builtin_amdgcn_wmma_f32_16x16x4_f32(
                false, A0, false, Bm, (short)0, c0, false, false);
        v8f c1 = {};
        c1 = __builtin_amdgcn_wmma_f32_16x16x4_f32(
                false, A1, false, Bm, (short)0, c1, false, false);

        float colm = 3.0e38f;
#pragma unroll
        for (int v = 0; v < 8; ++v) {
            float d20 = fmaxf(xxv0[v] + yy_col - 2.0f * c0[v], 0.0f);
            float d21 = fmaxf(xxv1[v] + yy_col - 2.0f * c1[v], 0.0f);
            rowmin0[v] = fminf(rowmin0[v], d20);
            rowmin1[v] = fminf(rowmin1[v], d21);
            colm = fminf(colm, fminf(d20, d21));
        }
        // min over the 32 rows handled by this wave: fold the two halves
        colm = fminf(colm, __shfl_xor(colm, 16, 32));
        if (half == 0) {
            // nonnegative floats: uint-bit order == float order
            atomicMin(colmin + (size_t)b * NPTS + t * 16 + lr,
                      __float_as_uint(colm));
        }
    }

    // ---- row-min reduction across the 16 lanes of each half ----
#pragma unroll
    for (int v = 0; v < 8; ++v) {
        float m0 = rowmin0[v], m1 = rowmin1[v];
        m0 = fminf(m0, __shfl_xor(m0, 1, 32));
        m0 = fminf(m0, __shfl_xor(m0, 2, 32));
        m0 = fminf(m0, __shfl_xor(m0, 4, 32));
        m0 = fminf(m0, __shfl_xor(m0, 8, 32));
        m1 = fminf(m1, __shfl_xor(m1, 1, 32));
        m1 = fminf(m1, __shfl_xor(m1, 2, 32));
        m1 = fminf(m1, __shfl_xor(m1, 4, 32));
        m1 = fminf(m1, __shfl_xor(m1, 8, 32));
        rowmin0[v] = m0; rowmin1[v] = m1;
    }
    if (lr == 0) {            // lanes 0 (rows v) and 16 (rows v+8)
        float s = 0.0f;
#pragma unroll
        for (int v = 0; v < 8; ++v) s += rowmin0[v] + rowmin1[v];
        atomicAdd(acc + 0, s);
    }
}

// ---------------------------------------------------------------------------
// PCA smallest-eigenvector of symmetric 3x3 covariance of 16 LDS points.
// ---------------------------------------------------------------------------
__device__ inline void pca_normal(const float* __restrict__ lds_pts,
                                  const int* __restrict__ bi,
                                  float& nx, float& ny, float& nz) {
    float mx = 0, my = 0, mz = 0;
    float sxx = 0, sxy = 0, sxz = 0, syy = 0, syz = 0, szz = 0;
#pragma unroll
    for (int k = 0; k < KNORM; ++k) {
        const int j = bi[k] * 3;
        const float x = lds_pts[j], y = lds_pts[j + 1], z = lds_pts[j + 2];
        mx += x; my += y; mz += z;
        sxx += x * x; sxy += x * y; sxz += x * z;
        syy += y * y; syz += y * z; szz += z * z;
    }
    const float inv = 1.0f / (float)KNORM;
    mx *= inv; my *= inv; mz *= inv;
    const float a00 = sxx * inv - mx * mx;
    const float a01 = sxy * inv - mx * my;
    const float a02 = sxz * inv - mx * mz;
    const float a11 = syy * inv - my * my;
    const float a12 = syz * inv - my * mz;
    const float a22 = szz * inv - mz * mz;

    // smallest eigenvalue (trigonometric method)
    const float p1 = a01 * a01 + a02 * a02 + a12 * a12;
    const float q  = (a00 + a11 + a22) * (1.0f / 3.0f);
    const float d0 = a00 - q, d1 = a11 - q, d2 = a22 - q;
    const float p2 = d0 * d0 + d1 * d1 + d2 * d2 + 2.0f * p1;
    const float p  = sqrtf(fmaxf(p2 * (1.0f / 6.0f), 1e-30f));
    const float ip = 1.0f / p;
    const float b00 = d0 * ip, b01 = a01 * ip, b02 = a02 * ip;
    const float b11 = d1 * ip, b12 = a12 * ip, b22 = d2 * ip;
    float r = 0.5f * (b00 * (b11 * b22 - b12 * b12)
                    - b01 * (b01 * b22 - b12 * b02)
                    + b02 * (b01 * b12 - b11 * b02));
    r = fminf(1.0f, fmaxf(-1.0f, r));
    const float phi  = acosf(r) * (1.0f / 3.0f);
    const float lmin = q + 2.0f * p * cosf(phi + 2.0943951023931953f); // +2pi/3

    // eigenvector: largest cross product among rows of (A - lmin*I)
    const float r0x = a00 - lmin, r0y = a01, r0z = a02;
    const float r1x = a01, r1y = a11 - lmin, r1z = a12;
    const float r2x = a02, r2y = a12, r2z = a22 - lmin;

    float c0x = r0y * r1z - r0z * r1y, c0y = r0z * r1x - r0x * r1z,
          c0z = r0x * r1y - r0y * r1x;
    float c1x = r0y * r2z - r0z * r2y, c1y = r0z * r2x - r0x * r2z,
          c1z = r0x * r2y - r0y * r2x;
    float c2x = r1y * r2z - r1z * r2y, c2y = r1z * r2x - r1x * r2z,
          c2z = r1x * r2y - r1y * r2x;
    const float n0 = c0x * c0x + c0y * c0y + c0z * c0z;
    const float n1 = c1x * c1x + c1y * c1y + c1z * c1z;
    const float n2 = c2x * c2x + c2y * c2y + c2z * c2z;

    float vx = c0x, vy = c0y, vz = c0z, nn = n0;
    if (n1 > nn) { vx = c1x; vy = c1y; vz = c1z; nn = n1; }
    if (n2 > nn) { vx = c2x; vy = c2y; vz = c2z; nn = n2; }
    const float rn = rsqrtf(fmaxf(nn, 1e-30f));
    vx *= rn; vy *= rn; vz *= rn;
    // deterministic sign convention
    if (vz < 0.0f || (vz == 0.0f && (vy < 0.0f || (vy == 0.0f && vx < 0.0f)))) {
        vx = -vx; vy = -vy; vz = -vz;
    }
    nx = vx; ny = vy; nz = vz;
}

// ---------------------------------------------------------------------------
// Per-point brute-force 16-NN over LDS-cached clouds + repulsion + normals.
// Both clouds async-staged into LDS (48 KB of the 320 KB WGP budget).
// ---------------------------------------------------------------------------
__global__ __launch_bounds__(256) void knn_pca_kernel(
        const float* __restrict__ pred, const float* __restrict__ gt,
        float* __restrict__ acc) {
    __shared__ float sp[NPTS * 3];
    __shared__ float sg[NPTS * 3];
    __shared__ float red[256];

    const int blocksPerBatch = NPTS / 256;
    const int b     = blockIdx.x / blocksPerBatch;
    const int chunk = blockIdx.x % blocksPerBatch;
    const int tid   = threadIdx.x;
    const int n     = chunk * 256 + tid;

    const float* pb = pred + (size_t)b * NPTS * 3;
    const float* gb = gt   + (size_t)b * NPTS * 3;

    const unsigned sp_base = (unsigned)(unsigned long long)(void*)sp;
    const unsigned sg_base = (unsigned)(unsigned long long)(void*)sg;
#pragma unroll
    for (int it = 0; it < (NPTS * 3 * 4) / (256 * 16); ++it) {
        const int off = (it * 256 + tid) * 16;
        async_copy_b128(sp_base + (unsigned)off,
                        (unsigned long long)((const char*)pb + off));
        async_copy_b128(sg_base + (unsigned)off,
                        (unsigned long long)((const char*)gb + off));
    }
    wait_async_all();
    __syncthreads();

    float bd[KNORM];
    int   bi[KNORM];

    // ---------- 16-NN on pred (self included) ----------
    const float qx = sp[n * 3], qy = sp[n * 3 + 1], qz = sp[n * 3 + 2];
#pragma unroll
    for (int k = 0; k < KNORM; ++k) { bd[k] = 3.0e38f; bi[k] = 0; }
    for (int j = 0; j < NPTS; ++j) {
        const float dx = sp[j * 3] - qx;
        const float dy = sp[j * 3 + 1] - qy;
        const float dz = sp[j * 3 + 2] - qz;
        float cd2 = dx * dx + dy * dy + dz * dz;
        int cidx = j;
        if (cd2 < bd[KNORM - 1]) {
#pragma unroll
            for (int k = 0; k < KNORM; ++k) {
                if (cd2 < bd[k]) {
                    const float td = bd[k]; const int ti = bi[k];
                    bd[k] = cd2; bi[k] = cidx;
                    cd2 = td; cidx = ti;
                }
            }
        }
    }

    // repulsion: entries 1..4 (entry 0 is self, d2=0)
    float rep = 0.0f;
#pragma unroll
    for (int k = 1; k <= KREP; ++k) {
        const float d = sqrtf(fmaxf(bd[k], 1e-12f));
        rep += fmaxf(0.02f - d, 0.0f);
    }

    float npx, npy, npz;
    pca_normal(sp, bi, npx, npy, npz);

    // ---------- 16-NN on gt (query = gt[n]) ----------
    const float hx = sg[n * 3], hy = sg[n * 3 + 1], hz = sg[n * 3 + 2];
#pragma unroll
    for (int k = 0; k < KNORM; ++k) { bd[k] = 3.0e38f; bi[k] = 0; }
    for (int j = 0; j < NPTS; ++j) {
        const float dx = sg[j * 3] - hx;
        const float dy = sg[j * 3 + 1] - hy;
        const float dz = sg[j * 3 + 2] - hz;
        float cd2 = dx * dx + dy * dy + dz * dz;
        int cidx = j;
        if (cd2 < bd[KNORM - 1]) {
#pragma unroll
            for (int k = 0; k < KNORM; ++k) {
                if (cd2 < bd[k]) {
                    const float td = bd[k]; const int ti = bi[k];
                    bd[k] = cd2; bi[k] = cidx;
                    cd2 = td; cidx = ti;
                }
            }
        }
    }

    float ngx, ngy, ngz;
    pca_normal(sg, bi, ngx, ngy, ngz);

    const float dotn = npx * ngx + npy * ngy + npz * ngz;

    // ---------- block reduction: rep, then dot ----------
    red[tid] = rep;
    __syncthreads();
    for (int w = 128; w > 0; w >>= 1) {
        if (tid < w) red[tid] += red[tid + w];
        __syncthreads();
    }
    if (tid == 0) atomicAdd(acc + 1, red[0]);
    __syncthreads();

    red[tid] = dotn;
    __syncthreads();
    for (int w = 128; w > 0; w >>= 1) {
        if (tid < w) red[tid] += red[tid + w];
        __syncthreads();
    }
    if (tid == 0) atomicAdd(acc + 2, red[0]);
}

// ---------------------------------------------------------------------------
// finalize: reduce colmin, combine weighted terms, write scalar.
// ---------------------------------------------------------------------------
__global__ __launch_bounds__(256) void finalize_kernel(
        const unsigned int* __restrict__ colmin,
        const float* __restrict__ acc, float* __restrict__ out) {
    __shared__ float red[256];
    const int tid = threadIdx.x;
    float s = 0.0f;
    for (int i = tid; i < BATCH * NPTS; i += 256)
        s += __uint_as_float(colmin[i]);
    red[tid] = s;
    __syncthreads();
    for (int w = 128; w > 0; w >>= 1) {
        if (tid < w) red[tid] += red[tid + w];
        __syncthreads();
    }
    if (tid == 0) {
        const float invBN = 1.0f / (float)(BATCH * NPTS);
        const float cd    = acc[0] * invBN + red[0] * invBN;
        const float rep   = acc[1] * (invBN / (float)KREP);
        const float normc = 1.0f - acc[2] * invBN;
        out[0] = 1.0f * cd + 0.1f * rep + 0.01f * normc;
    }
}

// ---------------------------------------------------------------------------
extern "C" void kernel_launch(void* const* d_in, const int* in_sizes, int n_in,
                              void* d_out, int out_size, void* d_ws,
                              size_t ws_size, hipStream_t stream) {
    const float* pred = (const float*)d_in[0];
    const float* gt   = (const float*)d_in[1];

    unsigned int* colmin = (unsigned int*)d_ws;                       // B*N uints
    float* acc = (float*)((char*)d_ws + (size_t)BATCH * NPTS * sizeof(unsigned int));
    float* out = (float*)d_out;

    init_kernel<<<(BATCH * NPTS + 255) / 256, 256, 0, stream>>>(colmin, acc);
    chamfer_wmma_kernel<<<BATCH * 8, 256, 0, stream>>>(pred, gt, colmin, acc);
    knn_pca_kernel<<<BATCH * (NPTS / 256), 256, 0, stream>>>(pred, gt, acc);
    finalize_kernel<<<1, 256, 0, stream>>>(colmin, acc, out);
}